// AttentionDecoder_56745107915056
// MI455X (gfx1250) — compile-verified
//
#include <hip/hip_runtime.h>
#include <hip/hip_bf16.h>

// Problem sizes (compile-time, match reference)
#define B_  64
#define S_  400
#define H_  512
#define E_  300
#define V_  50000
#define G4H (4 * H_)          // 2048
#define CAT2H (2 * H_)        // 1024

typedef float v2f __attribute__((ext_vector_type(2)));
typedef float v8f __attribute__((ext_vector_type(8)));

__device__ __forceinline__ float sigf(float x) { return 1.0f / (1.0f + __expf(-x)); }

// Load one 2-VGPR f32 fragment for V_WMMA_F32_16X16X4_F32.
// A (16x4, rows = matrix rows) and B (4x16, "rows" = the 16 N-columns of W,
// since B[k][n] = W[n][k]) share the same per-lane pattern:
//   lane L<16 : row = row0+L,     cols (k, k+1)
//   lane L>=16: row = row0+L-16,  cols (k+2, k+3)
__device__ __forceinline__ v2f ldfrag(const float* base, int ld, int row0, int k) {
  int lane = threadIdx.x & 31;
  int r = row0 + (lane & 15);
  int kk = k + ((lane >> 4) << 1);
  const float* p = base + (size_t)r * (size_t)ld + kk;
  v2f v; v.x = p[0]; v.y = p[1];
  return v;
}

__device__ __forceinline__ v8f wmma_f32(v2f a, v2f b, v8f c) {
  return __builtin_amdgcn_wmma_f32_16x16x4_f32(false, a, false, b, (short)0, c,
                                               false, false);
}

// ---------------- K1: embedding gather ----------------
__global__ void k_embed(const int* __restrict__ x, const float* __restrict__ emb,
                        float* __restrict__ xe) {
  int b = blockIdx.x;
  int row = x[b];
  for (int e = threadIdx.x; e < E_; e += blockDim.x)
    xe[b * E_ + e] = emb[(size_t)row * E_ + e];
}

// ---------------- K2: Bahdanau attention (scores, softmax, ctx) ----------------
__global__ void k_attn(const float* __restrict__ enc, const float* __restrict__ h0,
                       const float* __restrict__ Vw, const float* __restrict__ Vb,
                       float* __restrict__ w_attn, float* __restrict__ ctx) {
  int b = blockIdx.x;
  __shared__ float sh_h[H_];
  __shared__ float sh_vw[H_];
  __shared__ float sh_sc[S_];
  __shared__ float sred[8];
  int tid = threadIdx.x;                 // 256 threads = 8 waves
  int wave = tid >> 5, lane = tid & 31;

  for (int i = tid; i < H_; i += 256) { sh_h[i] = h0[b * H_ + i]; sh_vw[i] = Vw[i]; }
  __syncthreads();

  const float* encb = enc + (size_t)b * S_ * H_;
  // scores: score[s] = V_w . tanh(enc[b,s,:] + h_prev[b,:]) + V_b
  for (int s = wave; s < S_; s += 8) {
    const float* row = encb + (size_t)s * H_;
    float acc = 0.f;
    for (int j = lane; j < H_; j += 32)
      acc += tanhf(row[j] + sh_h[j]) * sh_vw[j];
    for (int off = 16; off; off >>= 1) acc += __shfl_xor(acc, off, 32);
    if (lane == 0) sh_sc[s] = acc + Vb[0];
  }
  __syncthreads();

  // softmax over S
  float m = -3.0e38f;
  for (int s = tid; s < S_; s += 256) m = fmaxf(m, sh_sc[s]);
  for (int off = 16; off; off >>= 1) m = fmaxf(m, __shfl_xor(m, off, 32));
  if (lane == 0) sred[wave] = m;
  __syncthreads();
  m = sred[0];
  for (int i = 1; i < 8; ++i) m = fmaxf(m, sred[i]);

  float sum = 0.f;
  for (int s = tid; s < S_; s += 256) {
    float e = __expf(sh_sc[s] - m);
    sh_sc[s] = e;
    sum += e;
  }
  for (int off = 16; off; off >>= 1) sum += __shfl_xor(sum, off, 32);
  __syncthreads();
  if (lane == 0) sred[wave] = sum;
  __syncthreads();
  sum = 0.f;
  for (int i = 0; i < 8; ++i) sum += sred[i];
  float inv = 1.0f / sum;
  for (int s = tid; s < S_; s += 256) {
    float wv = sh_sc[s] * inv;
    sh_sc[s] = wv;
    w_attn[b * S_ + s] = wv;
  }
  __syncthreads();

  // ctx[h] = sum_s w[s] * enc[b,s,h]  (coalesced over h)
  for (int h = tid; h < H_; h += 256) {
    float a = 0.f;
    for (int s = 0; s < S_; ++s) a += sh_sc[s] * encb[(size_t)s * H_ + h];
    ctx[b * H_ + h] = a;
  }
}

// ---------------- K3: LSTM gates GEMM via WMMA f32 16x16x4 ----------------
// gates[64,2048] = [xe|ctx] @ W_ih.T + h_prev @ W_hh.T + b_ih + b_hh
// grid = 128 blocks (N tiles of 16), block = 32 threads (1 wave), M = 4 tiles.
__global__ void k_gates(const float* __restrict__ xe, const float* __restrict__ ctx,
                        const float* __restrict__ h0, const float* __restrict__ Wih,
                        const float* __restrict__ Whh, const float* __restrict__ bih,
                        const float* __restrict__ bhh, float* __restrict__ gates) {
  int n0 = blockIdx.x * 16;
  v8f acc[4] = {};

  // K-span 1: xe (cols 0..299 of W_ih)
  for (int k = 0; k < E_; k += 4) {
    v2f bf = ldfrag(Wih, E_ + H_, n0, k);
#pragma unroll
    for (int mtile = 0; mtile < 4; ++mtile) {
      v2f af = ldfrag(xe, E_, mtile * 16, k);
      acc[mtile] = wmma_f32(af, bf, acc[mtile]);
    }
  }
  // K-span 2: ctx (cols 300..811 of W_ih)
  for (int k = 0; k < H_; k += 4) {
    v2f bf = ldfrag(Wih, E_ + H_, n0, E_ + k);
#pragma unroll
    for (int mtile = 0; mtile < 4; ++mtile) {
      v2f af = ldfrag(ctx, H_, mtile * 16, k);
      acc[mtile] = wmma_f32(af, bf, acc[mtile]);
    }
  }
  // K-span 3: h_prev @ W_hh.T
  for (int k = 0; k < H_; k += 4) {
    v2f bf = ldfrag(Whh, H_, n0, k);
#pragma unroll
    for (int mtile = 0; mtile < 4; ++mtile) {
      v2f af = ldfrag(h0, H_, mtile * 16, k);
      acc[mtile] = wmma_f32(af, bf, acc[mtile]);
    }
  }

  int lane = threadIdx.x & 31;
  int col = n0 + (lane & 15);
  float bias = bih[col] + bhh[col];
#pragma unroll
  for (int mtile = 0; mtile < 4; ++mtile)
#pragma unroll
    for (int j = 0; j < 8; ++j) {
      int r = mtile * 16 + j + ((lane >> 4) << 3);
      gates[r * G4H + col] = acc[mtile][j] + bias;
    }
}

// ---------------- K4: LSTM cell + p_gen + out_cat ----------------
__global__ void k_cell(const float* __restrict__ gates, const float* __restrict__ c0,
                       const float* __restrict__ ctx, const float* __restrict__ xe,
                       const float* __restrict__ genw, const float* __restrict__ genb,
                       float* __restrict__ out_h, float* __restrict__ out_c,
                       float* __restrict__ out_cat, float* __restrict__ p_gen) {
  int b = blockIdx.x, tid = threadIdx.x;   // 256 threads
  int wave = tid >> 5, lane = tid & 31;
  __shared__ float sh_h[H_];
  __shared__ float sred[8];

  for (int h = tid; h < H_; h += 256) {
    float gi = gates[b * G4H + h];
    float gf = gates[b * G4H + H_ + h];
    float gg = gates[b * G4H + 2 * H_ + h];
    float go = gates[b * G4H + 3 * H_ + h];
    float c = sigf(gf) * c0[b * H_ + h] + sigf(gi) * tanhf(gg);
    float hn = sigf(go) * tanhf(c);
    out_c[b * H_ + h] = c;
    out_h[b * H_ + h] = hn;
    sh_h[h] = hn;
    out_cat[b * CAT2H + h] = hn;
    out_cat[b * CAT2H + H_ + h] = ctx[b * H_ + h];
  }
  __syncthreads();

  // p_gen = sigmoid([ctx, h_new, xe] . gen_w + gen_b)
  float acc = 0.f;
  for (int i = tid; i < 2 * H_ + E_; i += 256) {
    float v = (i < H_) ? ctx[b * H_ + i]
            : (i < 2 * H_) ? sh_h[i - H_]
                           : xe[b * E_ + (i - 2 * H_)];
    acc += v * genw[i];
  }
  for (int off = 16; off; off >>= 1) acc += __shfl_xor(acc, off, 32);
  if (lane == 0) sred[wave] = acc;
  __syncthreads();
  if (tid == 0) {
    float t = 0.f;
    for (int i = 0; i < 8; ++i) t += sred[i];
    p_gen[b] = sigf(t + genb[0]);
  }
}

// ---------------- K5: big GEMM via WMMA f32 16x16x4 ----------------
// out[b,v] = p_gen[b] * (out_cat[b,:] . out_w[v,:] + out_b[v])
// grid = 3125 N-tiles, block = 32 threads, M = 64 (4 tiles) -> out_w read once.
__global__ void k_outgemm(const float* __restrict__ cat, const float* __restrict__ outw,
                          const float* __restrict__ outb, const float* __restrict__ p_gen,
                          float* __restrict__ out) {
  int n0 = blockIdx.x * 16;
  int lane = threadIdx.x & 31;
  v8f acc[4] = {};

  for (int k = 0; k < CAT2H; k += 4) {
    if (k + 64 < CAT2H)  // stream-prefetch out_w rows (global_prefetch_b8)
      __builtin_prefetch(outw + (size_t)(n0 + (lane & 15)) * CAT2H + k + 64, 0, 1);
    v2f bf = ldfrag(outw, CAT2H, n0, k);
#pragma unroll
    for (int mtile = 0; mtile < 4; ++mtile) {
      v2f af = ldfrag(cat, CAT2H, mtile * 16, k);
      acc[mtile] = wmma_f32(af, bf, acc[mtile]);
    }
  }

  int col = n0 + (lane & 15);
  float bias = outb[col];
#pragma unroll
  for (int mtile = 0; mtile < 4; ++mtile)
#pragma unroll
    for (int j = 0; j < 8; ++j) {
      int r = mtile * 16 + j + ((lane >> 4) << 3);
      out[(size_t)r * V_ + col] = p_gen[r] * (acc[mtile][j] + bias);
    }
}

// ---------------- K6: deterministic pointer scatter ----------------
// out[b, text[b,s]] = p_gen*(dot + out_b) + (1-p_gen)*w[b,s]; sequential s => last wins.
__global__ void k_scatter(const int* __restrict__ text, const float* __restrict__ w_attn,
                          const float* __restrict__ cat, const float* __restrict__ outw,
                          const float* __restrict__ outb, const float* __restrict__ p_gen,
                          float* __restrict__ out) {
  int b = blockIdx.x;
  int lane = threadIdx.x;   // 32 threads = 1 wave
  float pg = p_gen[b];
  float pp = 1.0f - pg;
  for (int s = 0; s < S_; ++s) {
    int t = text[b * S_ + s];
    float acc = 0.f;
    for (int k = lane; k < CAT2H; k += 32)
      acc += cat[b * CAT2H + k] * outw[(size_t)t * CAT2H + k];
    for (int off = 16; off; off >>= 1) acc += __shfl_xor(acc, off, 32);
    if (lane == 0)
      out[(size_t)b * V_ + t] = pg * (acc + outb[t]) + pp * w_attn[b * S_ + s];
  }
}

extern "C" void kernel_launch(void* const* d_in, const int* in_sizes, int n_in,
                              void* d_out, int out_size, void* d_ws, size_t ws_size,
                              hipStream_t stream) {
  const int*   x     = (const int*)d_in[0];
  const float* enc   = (const float*)d_in[1];
  const float* h0    = (const float*)d_in[2];   // layer 0 == base
  const float* c0    = (const float*)d_in[3];
  const int*   text  = (const int*)d_in[4];
  // d_in[5] = batch_size (compile-time 64)
  const float* emb   = (const float*)d_in[6];
  const float* Vw    = (const float*)d_in[7];
  const float* Vb    = (const float*)d_in[8];
  const float* genw  = (const float*)d_in[9];
  const float* genb  = (const float*)d_in[10];
  const float* outw  = (const float*)d_in[11];
  const float* outb  = (const float*)d_in[12];
  const float* Wih   = (const float*)d_in[13];
  const float* Whh   = (const float*)d_in[14];
  const float* bih   = (const float*)d_in[15];
  const float* bhh   = (const float*)d_in[16];

  float* out_prob = (float*)d_out;                         // [64,50000]
  float* out_h    = out_prob + (size_t)B_ * V_;            // [1,64,512]
  float* out_c    = out_h + (size_t)B_ * H_;               // [1,64,512]

  // Workspace layout (floats)
  float* ws      = (float*)d_ws;
  float* w_attn  = ws;                                     // 64*400
  float* ctx     = w_attn + B_ * S_;                       // 64*512
  float* xe      = ctx + B_ * H_;                          // 64*300
  float* gates   = xe + B_ * E_;                           // 64*2048
  float* out_cat = gates + B_ * G4H;                       // 64*1024
  float* p_gen   = out_cat + B_ * CAT2H;                   // 64

  k_embed<<<B_, 256, 0, stream>>>(x, emb, xe);
  k_attn<<<B_, 256, 0, stream>>>(enc, h0, Vw, Vb, w_attn, ctx);
  k_gates<<<G4H / 16, 32, 0, stream>>>(xe, ctx, h0, Wih, Whh, bih, bhh, gates);
  k_cell<<<B_, 256, 0, stream>>>(gates, c0, ctx, xe, genw, genb,
                                 out_h, out_c, out_cat, p_gen);
  k_outgemm<<<V_ / 16, 32, 0, stream>>>(out_cat, outw, outb, p_gen, out_prob);
  k_scatter<<<B_, 32, 0, stream>>>(text, w_attn, out_cat, outw, outb, p_gen, out_prob);
}